// SoftSkeletonLoss_63410897158821
// MI455X (gfx1250) — compile-verified
//
#include <hip/hip_runtime.h>

// SoftSkeletonLoss for (2,1,192,192,192) fp32 inputs.
//
// Mathematical reduction (exact, input-independent):
//   In _soft_skeletonize, x = sigmoid(.) >= 0. Wherever x(v)>0 the 3^3
//   ones-conv at v includes v => eroded(v)=1 => opened(v)=1 => clip(x-1,0,1)=0;
//   where x(v)==0 the skeleton is already 0. So skeleton == 0 after iteration 1
//   for ANY input => intersection=union=0 => skeleton_loss = 1-EPS/EPS = 0.
// Remaining work: connectivity loss from _count_components on
//   pred_binary = (sigmoid(pred) > 0.5)  (== pred > 0)  and raw float target:
//   nb = conv3x3x3(vol);  cnt = sum(vol>0);  S = sum_{vol>0}(nb + EPS)
//   C = cnt / (S / max(cnt,1));  out = |C_pred - C_target|
//
// Bandwidth-bound (~161MB reads => ~7us @ 23.3TB/s). One fused LDS-tiled pass
// (16^3 tile, 18^3 halo). Data movement: CDNA5 GLOBAL_LOAD_ASYNC_TO_LDS_B32
// (ASYNCcnt; signature (int* AS1, int* AS3, imm off, imm cpol)) — global->LDS
// direct, no VGPR round-trip, ~46 transfers in flight/thread, one
// s_wait_asynccnt 0 before the barrier. OOB halo cells zeroed via plain ds
// stores (disjoint addresses from async writes => no cross-counter race).
// Pred tile is then binarized IN PLACE (reg-staged, double barrier) so the
// 162 stencil reads/thread need no per-read threshold (-~300 VALU/thread,
// keeps us near the HBM roofline instead of VALU-bound).
// Wave-level sums use V_WMMA_F32_16X16X4_F32 with an all-ones B matrix:
// every A element lands in exactly one row-sum, so sum_m D[m,n0] equals the
// 32-lane total independent of A's element->(m,k) mapping. C/D layout: lane
// n=lane&15 holds rows m = vgpr + 8*(lane>>4) => total = (per-lane sum of the
// 8 accumulator VGPRs) at lane 0 + lane 16. Deterministic two-stage final
// reduction (fixed-order, no float atomics).

#define DIMV 192
#define TILE 16
#define HALO 18
#define HHH  (HALO * HALO * HALO)       // 5832
#define TPB  256
#define NELT 23                          // ceil(HHH / TPB)
#define TPD  (DIMV / TILE)              // 12
#define TILES_PER_IMG (TPD * TPD * TPD) // 1728
#define NIMG 2
#define NBLK (NIMG * TILES_PER_IMG)     // 3456
#define EPSF 1e-8f

#if defined(__gfx1250__) && __has_builtin(__builtin_amdgcn_global_load_async_to_lds_b32)
#define USE_ASYNC_LDS 1
#else
#define USE_ASYNC_LDS 0
#endif

typedef __attribute__((ext_vector_type(2))) float v2f;
typedef __attribute__((ext_vector_type(8))) float v8f;
#if USE_ASYNC_LDS
typedef __attribute__((address_space(1))) int as1i;  // global (AS1) int
typedef __attribute__((address_space(3))) int as3i;  // LDS (AS3) int
#endif

__device__ __forceinline__ float wave_sum_wmma(float v) {
  // A tile: this lane contributes (v, 0); B = all ones (4x16).
  v2f a; a[0] = v;    a[1] = 0.0f;
  v2f b; b[0] = 1.0f; b[1] = 1.0f;
  v8f c = {};
  c = __builtin_amdgcn_wmma_f32_16x16x4_f32(
      /*neg_a=*/false, a, /*neg_b=*/false, b,
      /*c_mod=*/(short)0, c, /*reuse_a=*/false, /*reuse_b=*/false);
  float s = c[0] + c[1] + c[2] + c[3] + c[4] + c[5] + c[6] + c[7];
  // rows 0..7 live in lanes 0..15, rows 8..15 in lanes 16..31 (column n=lane&15)
  return __shfl(s, 0, 32) + __shfl(s, 16, 32);
}

__global__ __launch_bounds__(TPB)
void skel_tile_kernel(const float* __restrict__ pred,
                      const float* __restrict__ target,
                      float* __restrict__ partials) {
  __shared__ float sp[HALO][HALO][HALO];  // pred: raw, then binarized in place
  __shared__ float st[HALO][HALO][HALO];  // raw target
  __shared__ float wred[8][4];

  const int blk = blockIdx.x;
  const int img = blk / TILES_PER_IMG;
  const int t3  = blk % TILES_PER_IMG;
  const int tz  = t3 / (TPD * TPD);
  const int ty  = (t3 / TPD) % TPD;
  const int tx  = t3 % TPD;
  const long base = (long)img * DIMV * DIMV * DIMV;
  const int ox = tx * TILE - 1, oy = ty * TILE - 1, oz = tz * TILE - 1;

  const int tid = threadIdx.x;

#if USE_ASYNC_LDS
  // low 32 bits of a generic pointer to LDS == AS(3) byte offset (flat rules)
  const unsigned sp_base = (unsigned)(unsigned long long)(void*)&sp[0][0][0];
  const unsigned st_base = (unsigned)(unsigned long long)(void*)&st[0][0][0];
#endif

  // ---- load 18^3 halo tile (zero padding outside volume) ----
  for (int e = tid; e < HHH; e += TPB) {
    int lx = e % HALO;
    int ly = (e / HALO) % HALO;
    int lz = e / (HALO * HALO);
    int gx = ox + lx, gy = oy + ly, gz = oz + lz;
    bool inb = (unsigned)gx < DIMV && (unsigned)gy < DIMV && (unsigned)gz < DIMV;
#if USE_ASYNC_LDS
    if (inb) {
      long gi = base + ((long)gz * DIMV + gy) * DIMV + gx;
      unsigned loff = (unsigned)e * 4u;
      __builtin_amdgcn_global_load_async_to_lds_b32(
          (as1i*)(unsigned long long)(pred + gi),
          (as3i*)(unsigned long long)(sp_base + loff), 0, 0);
      __builtin_amdgcn_global_load_async_to_lds_b32(
          (as1i*)(unsigned long long)(target + gi),
          (as3i*)(unsigned long long)(st_base + loff), 0, 0);
    } else {
      (&sp[0][0][0])[e] = 0.0f;   // disjoint from async dsts: no race
      (&st[0][0][0])[e] = 0.0f;
    }
#else
    float pv = 0.0f, tv = 0.0f;
    if (inb) {
      long gi = base + ((long)gz * DIMV + gy) * DIMV + gx;
      pv = pred[gi];
      tv = target[gi];
    }
    (&sp[0][0][0])[e] = pv;
    (&st[0][0][0])[e] = tv;
#endif
  }
#if USE_ASYNC_LDS
#if __has_builtin(__builtin_amdgcn_s_wait_asynccnt)
  __builtin_amdgcn_s_wait_asynccnt(0);
#else
  asm volatile("s_wait_asynccnt 0x0" ::: "memory");
#endif
#endif
  __syncthreads();

  // ---- binarize pred tile in place (raw > 0  <=>  sigmoid(raw) > 0.5) ----
  // Fixed-trip unrolled loops + static indexing => stays in registers.
  {
    float rawv[NELT];
#pragma unroll
    for (int i = 0; i < NELT; ++i) {
      int e = tid + i * TPB;
      rawv[i] = (e < HHH) ? (&sp[0][0][0])[e] : 0.0f;
    }
    __syncthreads();
#pragma unroll
    for (int i = 0; i < NELT; ++i) {
      int e = tid + i * TPB;
      if (e < HHH) (&sp[0][0][0])[e] = (rawv[i] > 0.0f) ? 1.0f : 0.0f;
    }
  }
  __syncthreads();

  // ---- each thread owns one (x,y) column of the 16^3 interior ----
  const int lx = (tid & 15) + 1;
  const int ly = (tid >> 4) + 1;

  // separable box sums: 3x3 xy-plane sums for all 18 z slices
  float pp[HALO], pt[HALO];
#pragma unroll
  for (int z = 0; z < HALO; ++z) {
    float a = 0.0f, b = 0.0f;
#pragma unroll
    for (int dy = -1; dy <= 1; ++dy)
#pragma unroll
      for (int dx = -1; dx <= 1; ++dx) {
        a += sp[z][ly + dy][lx + dx];   // already 0/1
        b += st[z][ly + dy][lx + dx];
      }
    pp[z] = a;
    pt[z] = b;
  }

  float cnt_p = 0.0f, S_p = 0.0f, cnt_t = 0.0f, S_t = 0.0f;
#pragma unroll
  for (int lz = 1; lz <= TILE; ++lz) {
    float bp  = sp[lz][ly][lx];          // 0/1
    float nbp = pp[lz - 1] + pp[lz] + pp[lz + 1];
    bool  mp  = bp > 0.0f;
    cnt_p += mp ? 1.0f : 0.0f;
    S_p   += mp ? (nbp + EPSF) : 0.0f;

    float bt  = st[lz][ly][lx];
    float nbt = pt[lz - 1] + pt[lz] + pt[lz + 1];
    bool  mt  = bt > 0.0f;
    cnt_t += mt ? 1.0f : 0.0f;
    S_t   += mt ? (nbt + EPSF) : 0.0f;
  }

  // ---- wave-level reduction via V_WMMA_F32_16X16X4_F32 (EXEC all ones) ----
  float r0 = wave_sum_wmma(cnt_p);
  float r1 = wave_sum_wmma(S_p);
  float r2 = wave_sum_wmma(cnt_t);
  float r3 = wave_sum_wmma(S_t);

  const int lane = tid & 31;
  const int wave = tid >> 5;
  if (lane == 0) {
    wred[wave][0] = r0; wred[wave][1] = r1;
    wred[wave][2] = r2; wred[wave][3] = r3;
  }
  __syncthreads();

  if (tid == 0) {
    float b0 = 0.0f, b1 = 0.0f, b2 = 0.0f, b3 = 0.0f;
#pragma unroll
    for (int w = 0; w < 8; ++w) {  // fixed order: deterministic
      b0 += wred[w][0]; b1 += wred[w][1];
      b2 += wred[w][2]; b3 += wred[w][3];
    }
    partials[blk * 4 + 0] = b0;
    partials[blk * 4 + 1] = b1;
    partials[blk * 4 + 2] = b2;
    partials[blk * 4 + 3] = b3;
  }
}

__global__ __launch_bounds__(TPB)
void finalize_kernel(const float* __restrict__ partials, int nblocks,
                     float* __restrict__ out) {
  __shared__ double red[TPB][4];
  double a0 = 0.0, a1 = 0.0, a2 = 0.0, a3 = 0.0;
  for (int i = threadIdx.x; i < nblocks; i += TPB) {
    a0 += (double)partials[i * 4 + 0];
    a1 += (double)partials[i * 4 + 1];
    a2 += (double)partials[i * 4 + 2];
    a3 += (double)partials[i * 4 + 3];
  }
  red[threadIdx.x][0] = a0; red[threadIdx.x][1] = a1;
  red[threadIdx.x][2] = a2; red[threadIdx.x][3] = a3;
  __syncthreads();
  for (int s = TPB / 2; s > 0; s >>= 1) {
    if ((int)threadIdx.x < s) {
      red[threadIdx.x][0] += red[threadIdx.x + s][0];
      red[threadIdx.x][1] += red[threadIdx.x + s][1];
      red[threadIdx.x][2] += red[threadIdx.x + s][2];
      red[threadIdx.x][3] += red[threadIdx.x + s][3];
    }
    __syncthreads();
  }
  if (threadIdx.x == 0) {
    double cnt_p = red[0][0], S_p = red[0][1];
    double cnt_t = red[0][2], S_t = red[0][3];
    double mean_p = S_p / fmax(cnt_p, 1.0);
    double C_p = cnt_p / mean_p;
    double mean_t = S_t / fmax(cnt_t, 1.0);
    double C_t = cnt_t / mean_t;
    double connectivity = fabs(C_p - C_t);
    // skeleton_loss == 0 exactly (see header proof); loss = ALPHA*0 + BETA*conn
    out[0] = (float)connectivity;
  }
}

extern "C" void kernel_launch(void* const* d_in, const int* in_sizes, int n_in,
                              void* d_out, int out_size, void* d_ws, size_t ws_size,
                              hipStream_t stream) {
  (void)in_sizes; (void)n_in; (void)out_size; (void)ws_size;
  const float* pred   = (const float*)d_in[0];
  const float* target = (const float*)d_in[1];
  float* partials = (float*)d_ws;          // NBLK*4 floats = 55,296 B
  float* out = (float*)d_out;

  skel_tile_kernel<<<dim3(NBLK), dim3(TPB), 0, stream>>>(pred, target, partials);
  finalize_kernel<<<dim3(1), dim3(TPB), 0, stream>>>(partials, NBLK, out);
}